// DeepWalk_13718125543770
// MI455X (gfx1250) — compile-verified
//
#include <hip/hip_runtime.h>
#include <hip/hip_bf16.h>
#include <math.h>

// DeepWalk hierarchical-softmax loss on MI455X (gfx1250, wave32).
// One wave per edge; V_WMMA_F32_16X16X4_F32 computes the 20 path-row dot
// products (zv broadcast across the N dimension keeps fp32 precision).
// Memory-bound (~350 MB gathered @ 23.3 TB/s); this revision removes the
// instruction-issue bottleneck (16 libm log_sigmoid chains -> 1 fast chain).

#define D_DIM 128   // embedding dim (reference constant)
#define L_PATH 20   // max path length (reference constant)
#define WAVES_PER_BLOCK 8

typedef float v2f __attribute__((ext_vector_type(2)));
typedef float v8f __attribute__((ext_vector_type(8)));

__global__ __launch_bounds__(256) void hs_loss_kernel(
    const int* __restrict__ edges,       // (B,2) int32
    const int* __restrict__ path_nodes,  // (N,L) int32
    const int* __restrict__ path_signs,  // (N,L) int32 in {-1,0,1}; 0 == masked
    const float* __restrict__ Z,         // (2N-1, D) fp32
    float* __restrict__ partial,         // one partial sum per block
    int B)
{
  __shared__ float zv_s[WAVES_PER_BLOCK][D_DIM];
  __shared__ float red_s[WAVES_PER_BLOCK];

  const int tid  = threadIdx.x;
  const int lane = tid & 31;
  const int wave = tid >> 5;

  int e = blockIdx.x * WAVES_PER_BLOCK + wave;
  const bool ve = (e < B);
  if (!ve) e = 0;                       // keep EXEC full for WMMA; zero acc later

  const int ctx  = edges[2 * e + 0];
  const int leaf = edges[2 * e + 1];

  // Stage zv = Z[ctx] into LDS: 128 floats, 4 per lane, coalesced b128.
  {
    const float4 v = ((const float4*)(Z + (size_t)ctx * D_DIM))[lane];
    ((float4*)zv_s[wave])[lane] = v;
  }

  // Lane l (< 20) preloads sign for path position l; padding lanes hold 0.
  float sign_own = 0.0f;
  if (lane < L_PATH)
    sign_own = (float)path_signs[(size_t)leaf * L_PATH + lane];

  __syncthreads();   // zv_s visible across lanes (cross-lane LDS RAW)

  // WMMA f32 16x16x4 operand placement (wave32):
  //   A 16x4 : lane m=lane&15 is row m; V0 holds K=koff, V1 holds K=koff+1,
  //            koff = 0 (lanes 0-15) or 2 (lanes 16-31).
  //   B 4x16 : broadcast zv -> same K split per lane half.
  //   C/D    : VGPR j, lanes 0-15 -> row M=j; lanes 16-31 -> row M=j+8.
  const int koff = (lane >> 4) << 1;
  const int mrow = lane & 15;

  // Row gathers for both 16-row tiles (tile0: l=0..15, tile1: l=16..19+pad).
  const int node0 = path_nodes[(size_t)leaf * L_PATH + mrow];
  const int node1 = (mrow < L_PATH - 16)
                        ? path_nodes[(size_t)leaf * L_PATH + 16 + mrow] : 0;
  const float* a0row = Z + (size_t)node0 * D_DIM + koff;
  const float* a1row = Z + (size_t)node1 * D_DIM + koff;
  const float* brow  = &zv_s[wave][koff];

  // Fused K-loop: one LDS B load feeds both tiles; alternating c0/c1 WMMAs
  // give dependency distance 2 on each accumulator (better XDL pipelining).
  v8f c0 = {};
  v8f c1 = {};
#pragma unroll
  for (int k = 0; k < D_DIM / 4; ++k) {
    v2f b  = *(const v2f*)(brow + 4 * k);    // ds_load_b64 (broadcast)
    v2f a0 = *(const v2f*)(a0row + 4 * k);   // global_load_b64 (row-contiguous)
    v2f a1 = *(const v2f*)(a1row + 4 * k);
    c0 = __builtin_amdgcn_wmma_f32_16x16x4_f32(
        false, a0, false, b, (short)0, c0, false, false);
    c1 = __builtin_amdgcn_wmma_f32_16x16x4_f32(
        false, a1, false, b, (short)0, c1, false, false);
  }

  // Transpose x into lanes: row M of D is replicated across 16 lanes, so
  // pick xs = c[lane&7] (7-step cndmask tree, shared compare for c0/c1),
  // then one shuffle lands x_l in lane l for each tile.
  const int m7 = lane & 7;
  float xs0 = c0[0];
  float xs1 = c1[0];
#pragma unroll
  for (int j = 1; j < 8; ++j) {
    const bool sel = (m7 == j);
    xs0 = sel ? c0[j] : xs0;
    xs1 = sel ? c1[j] : xs1;
  }
  // lanes 0-15 of xs hold rows 0..7 (dup x2); lanes 16-31 hold rows 8..15.
  const int src = (lane & 7) | ((lane & 8) << 1);
  const float xr0 = __shfl(xs0, src, 32);   // lane l (<16): x[l]      (tile 0)
  const float xr1 = __shfl(xs1, src, 32);   // lane l (>=16): x[l]     (tile 1)
  const float x   = (lane < 16) ? xr0 : xr1;

  // Single stable log_sigmoid chain for the whole wave:
  //   log_sigmoid(y) = min(y,0) - log(1 + exp(-|y|))
  const float sx = sign_own * x;
  const float lp = fminf(sx, 0.0f) - __logf(1.0f + __expf(-fabsf(sx)));
  float acc = (ve && (sign_own != 0.0f)) ? lp : 0.0f;

  // Deterministic wave tree reduction, then block reduction in LDS.
#pragma unroll
  for (int o = 16; o > 0; o >>= 1) acc += __shfl_xor(acc, o, 32);
  if (lane == 0) red_s[wave] = acc;
  __syncthreads();
  if (tid == 0) {
    float tsum = 0.0f;
#pragma unroll
    for (int w = 0; w < WAVES_PER_BLOCK; ++w) tsum += red_s[w];
    partial[blockIdx.x] = tsum;
  }
}

// Single-block deterministic final reduction: out = -sum(partial).
__global__ __launch_bounds__(256) void hs_reduce_kernel(
    const float* __restrict__ partial, float* __restrict__ out, int n)
{
  __shared__ float red_s[8];
  float s = 0.0f;
  for (int i = threadIdx.x; i < n; i += 256) s += partial[i];
#pragma unroll
  for (int o = 16; o > 0; o >>= 1) s += __shfl_xor(s, o, 32);
  const int lane = threadIdx.x & 31;
  const int wave = threadIdx.x >> 5;
  if (lane == 0) red_s[wave] = s;
  __syncthreads();
  if (threadIdx.x == 0) {
    float t = 0.0f;
#pragma unroll
    for (int w = 0; w < 8; ++w) t += red_s[w];
    out[0] = -t;
  }
}

extern "C" void kernel_launch(void* const* d_in, const int* in_sizes, int n_in,
                              void* d_out, int out_size, void* d_ws, size_t ws_size,
                              hipStream_t stream) {
  const int*   edges      = (const int*)d_in[0];
  const int*   path_nodes = (const int*)d_in[1];
  const int*   path_signs = (const int*)d_in[2];
  // d_in[3] = path_mask (bool) -- redundant: mask <=> sign != 0
  const float* Z          = (const float*)d_in[4];
  float*       out        = (float*)d_out;
  float*       partial    = (float*)d_ws;

  const int B = in_sizes[0] / 2;
  const int nblocks = (B + WAVES_PER_BLOCK - 1) / WAVES_PER_BLOCK;  // 4096

  hipLaunchKernelGGL(hs_loss_kernel, dim3(nblocks), dim3(256), 0, stream,
                     edges, path_nodes, path_signs, Z, partial, B);
  hipLaunchKernelGGL(hs_reduce_kernel, dim3(1), dim3(256), 0, stream,
                     partial, out, nblocks);
}